// OT_80522046865605
// MI455X (gfx1250) — compile-verified
//
#include <hip/hip_runtime.h>
#include <cmath>

// Problem constants (from reference)
#define NUMN 8
#define SB 4
#define TB 4
#define DIMD 512
#define NPTS 1024   // H*W = 32*32
#define KC 64
#define KM_ITERS 10

typedef float v2f __attribute__((ext_vector_type(2)));
typedef float v8f __attribute__((ext_vector_type(8)));

// D = A(16x4 f32) * B(4x16 f32) + C(16x16 f32)   -- CDNA5 wave32 WMMA
__device__ __forceinline__ v8f wmma4(v2f a, v2f b, v8f c) {
    return __builtin_amdgcn_wmma_f32_16x16x4_f32(
        /*neg_a=*/false, a, /*neg_b=*/false, b,
        /*c_mod=*/(short)0, c, /*reuse_a=*/false, /*reuse_b=*/false);
}

// ---------------------------------------------------------------------------
// Kernel 1: deterministic k-means init: centers[k] = point k
// grid 64 (32 source runs + 32 target runs), block 256
// centers stored as [n][b][k][d] (already transposed to cs/ct layout)
// ---------------------------------------------------------------------------
__global__ void k_init_centers(const float* __restrict__ sfeat,
                               const float* __restrict__ tfeat,
                               float* __restrict__ cents) {
    int run = blockIdx.x;
    bool src = run < 32;
    int r = run & 31;
    int b = r >> 3, n = r & 7;
    const float* feat = (src ? sfeat : tfeat) + (size_t)(b * NUMN + n) * DIMD * NPTS;
    float* cent = cents + (src ? 0 : (size_t)SB * NUMN * KC * DIMD)
                        + (size_t)(n * SB + b) * KC * DIMD;
    for (int idx = threadIdx.x; idx < KC * DIMD; idx += blockDim.x) {
        int k = idx >> 9, d = idx & 511;
        cent[idx] = feat[(size_t)d * NPTS + k];
    }
}

// ---------------------------------------------------------------------------
// Kernel 2: assignment via WMMA.  score = |c|^2 - 2 p.c  ==> argmax of
// (p.c - 0.5|c|^2); the -0.5|c|^2 bias is folded into the accumulator init.
// grid 64, block 256 (8 waves); each wave owns 8 point-tiles of 16.
// ---------------------------------------------------------------------------
__global__ void k_assign(const float* __restrict__ sfeat,
                         const float* __restrict__ tfeat,
                         const float* __restrict__ cents,
                         int* __restrict__ assign_ws) {
    __shared__ float s_cn[KC];
    int run = blockIdx.x;
    bool src = run < 32;
    int r = run & 31;
    int b = r >> 3, n = r & 7;
    const float* feat = (src ? sfeat : tfeat) + (size_t)(b * NUMN + n) * DIMD * NPTS;
    const float* cent = cents + (src ? 0 : (size_t)SB * NUMN * KC * DIMD)
                              + (size_t)(n * SB + b) * KC * DIMD;
    int* assign = assign_ws + run * NPTS;

    for (int k = threadIdx.x; k < KC; k += blockDim.x) {
        float s = 0.f;
        for (int d = 0; d < DIMD; ++d) { float v = cent[k * DIMD + d]; s += v * v; }
        s_cn[k] = s;
    }
    __syncthreads();

    const int lane = threadIdx.x & 31;
    const int wave = threadIdx.x >> 5;
    const int m = lane & 15;
    const int kkb = (lane >> 4) << 1;   // 0 or 2

    for (int pt = wave; pt < NPTS / 16; pt += 8) {
        const int p0 = pt << 4;
        v8f acc[4];
#pragma unroll
        for (int c = 0; c < 4; ++c) {
            float init = -0.5f * s_cn[c * 16 + m];
#pragma unroll
            for (int i = 0; i < 8; ++i) acc[c][i] = init;
        }
        for (int k0 = 0; k0 < DIMD; k0 += 4) {
            v2f a;
            a.x = feat[(size_t)(k0 + kkb) * NPTS + p0 + m];
            a.y = feat[(size_t)(k0 + kkb + 1) * NPTS + p0 + m];
            __builtin_prefetch(&feat[(size_t)(k0 + 8 + kkb) * NPTS + p0 + m], 0, 0);
#pragma unroll
            for (int c = 0; c < 4; ++c) {
                v2f bb = *(const v2f*)&cent[(c * 16 + m) * DIMD + k0 + kkb];
                acc[c] = wmma4(a, bb, acc[c]);
            }
        }
        // argmax across 64 centers: 4 c-tiles in regs, 16 lanes per half-wave
#pragma unroll
        for (int rr = 0; rr < 8; ++rr) {
            float bv = acc[0][rr];
            int bi = m;
#pragma unroll
            for (int c = 1; c < 4; ++c) {
                float v = acc[c][rr];
                if (v > bv) { bv = v; bi = c * 16 + m; }
            }
#pragma unroll
            for (int off = 1; off < 16; off <<= 1) {
                float ov = __shfl_xor(bv, off, 32);
                int   oi = __shfl_xor(bi, off, 32);
                if (ov > bv || (ov == bv && oi < bi)) { bv = ov; bi = oi; }
            }
            if (m == 0) assign[p0 + rr + ((lane >> 4) << 3)] = bi;
        }
    }
}

// ---------------------------------------------------------------------------
// Kernel 3: cluster update. LDS histogram (64x512 f32 = 128KB, ds_add_f32).
// Empty clusters keep old center. grid 64, block 256, dynamic LDS.
// ---------------------------------------------------------------------------
__global__ void k_update(const float* __restrict__ sfeat,
                         const float* __restrict__ tfeat,
                         const int* __restrict__ assign_ws,
                         float* __restrict__ cents) {
    extern __shared__ float lds[];
    float* sums = lds;              // KC*DIMD
    float* cnt  = lds + KC * DIMD;  // KC
    int run = blockIdx.x;
    bool src = run < 32;
    int r = run & 31;
    int b = r >> 3, n = r & 7;
    const float* feat = (src ? sfeat : tfeat) + (size_t)(b * NUMN + n) * DIMD * NPTS;
    float* cent = cents + (src ? 0 : (size_t)SB * NUMN * KC * DIMD)
                        + (size_t)(n * SB + b) * KC * DIMD;
    const int* assign = assign_ws + run * NPTS;

    for (int i = threadIdx.x; i < KC * DIMD; i += blockDim.x) sums[i] = 0.f;
    for (int i = threadIdx.x; i < KC; i += blockDim.x) cnt[i] = 0.f;
    __syncthreads();

    for (int p = threadIdx.x; p < NPTS; p += blockDim.x) {
        int a = assign[p];
        atomicAdd(&cnt[a], 1.f);
        for (int d = 0; d < DIMD; ++d)
            atomicAdd(&sums[a * DIMD + d], feat[(size_t)d * NPTS + p]);
    }
    __syncthreads();

    for (int idx = threadIdx.x; idx < KC * DIMD; idx += blockDim.x) {
        float c = cnt[idx >> 9];
        if (c > 0.f) cent[idx] = sums[idx] / c;
    }
}

// ---------------------------------------------------------------------------
// Kernel 4: y[n,i,j,k] = argmin_l dist(cs[n,i,k], ct[n,j,l])  via WMMA dots.
// grid 128 = (n,i,j), block 128 (4 waves); wave w owns k-tile w.
// ---------------------------------------------------------------------------
__global__ void k_crossassign(const float* __restrict__ cs,
                              const float* __restrict__ ct,
                              int* __restrict__ y_ws) {
    __shared__ float s_tn[KC];
    int bid = blockIdx.x;
    int n = bid >> 4, i = (bid >> 2) & 3, j = bid & 3;
    const float* A = cs + (size_t)(n * SB + i) * KC * DIMD;
    const float* B = ct + (size_t)(n * TB + j) * KC * DIMD;
    int* y = y_ws + bid * KC;

    for (int l = threadIdx.x; l < KC; l += blockDim.x) {
        float s = 0.f;
        for (int d = 0; d < DIMD; ++d) { float v = B[l * DIMD + d]; s += v * v; }
        s_tn[l] = s;
    }
    __syncthreads();

    const int lane = threadIdx.x & 31;
    const int wave = threadIdx.x >> 5;
    const int m = lane & 15;
    const int kkb = (lane >> 4) << 1;
    const int k0 = wave << 4;

    v8f acc[4];
#pragma unroll
    for (int c = 0; c < 4; ++c) {
        float init = -0.5f * s_tn[c * 16 + m];
#pragma unroll
        for (int q = 0; q < 8; ++q) acc[c][q] = init;
    }
    for (int d0 = 0; d0 < DIMD; d0 += 4) {
        v2f a = *(const v2f*)&A[(k0 + m) * DIMD + d0 + kkb];
#pragma unroll
        for (int c = 0; c < 4; ++c) {
            v2f bb = *(const v2f*)&B[(c * 16 + m) * DIMD + d0 + kkb];
            acc[c] = wmma4(a, bb, acc[c]);
        }
    }
#pragma unroll
    for (int rr = 0; rr < 8; ++rr) {
        float bv = acc[0][rr];
        int bi = m;
#pragma unroll
        for (int c = 1; c < 4; ++c) {
            float v = acc[c][rr];
            if (v > bv) { bv = v; bi = c * 16 + m; }
        }
#pragma unroll
        for (int off = 1; off < 16; off <<= 1) {
            float ov = __shfl_xor(bv, off, 32);
            int   oi = __shfl_xor(bi, off, 32);
            if (ov > bv || (ov == bv && oi < bi)) { bv = ov; bi = oi; }
        }
        if (m == 0) y[k0 + rr + ((lane >> 4) << 3)] = bi;
    }
}

// ---------------------------------------------------------------------------
// Kernel 5: per n: STT = S^T T  and  G = T^T T  (both 512x512, N=1024 rows),
// with S/T rows resolved through precomputed row-offset tables in LDS.
// grid 8*128, block 256 (one 16x16 tile per wave, two accumulators).
// ---------------------------------------------------------------------------
__global__ void k_gram(const float* __restrict__ cs,
                       const float* __restrict__ ct,
                       const int* __restrict__ y_ws,
                       float* __restrict__ STT,
                       float* __restrict__ G) {
    __shared__ int srow[NPTS];
    __shared__ int trow[NPTS];
    int n = blockIdx.x >> 7, tg = blockIdx.x & 127;
    for (int r = threadIdx.x; r < NPTS; r += blockDim.x) {
        int i = r >> 8, j = (r >> 6) & 3, k = r & 63;
        srow[r] = ((n * SB + i) * KC + k) * DIMD;
        int yv = y_ws[((n * SB + i) * TB + j) * KC + k];
        trow[r] = ((n * TB + j) * KC + yv) * DIMD;
    }
    __syncthreads();

    const int lane = threadIdx.x & 31;
    const int wave = threadIdx.x >> 5;
    const int t = tg * 8 + wave;                 // 0..1023 tiles
    const int d0 = (t >> 5) << 4, e0 = (t & 31) << 4;
    const int m = lane & 15;
    const int kkb = (lane >> 4) << 1;

    v8f accS = {};
    v8f accG = {};
    for (int r0 = 0; r0 < NPTS; r0 += 4) {
        int sa = srow[r0 + kkb],  sb2 = srow[r0 + kkb + 1];
        int ta = trow[r0 + kkb],  tb2 = trow[r0 + kkb + 1];
        v2f as; as.x = cs[sa + d0 + m]; as.y = cs[sb2 + d0 + m];
        v2f at; at.x = ct[ta + d0 + m]; at.y = ct[tb2 + d0 + m];
        v2f bt; bt.x = ct[ta + e0 + m]; bt.y = ct[tb2 + e0 + m];
        accS = wmma4(as, bt, accS);
        accG = wmma4(at, bt, accG);
    }
    size_t base = (size_t)n * DIMD * DIMD;
#pragma unroll
    for (int rr = 0; rr < 8; ++rr) {
        int row = d0 + rr + ((lane >> 4) << 3);
        int col = e0 + m;
        STT[base + row * DIMD + col] = accS[rr];
        G  [base + row * DIMD + col] = accG[rr];
    }
}

// ---------------------------------------------------------------------------
// Kernel 6: Gauss-Jordan inverse of SPD G (diagonal pivoting is valid for SPD).
// grid 8 (one block per n), block 256, 512 sequential pivots.
// ---------------------------------------------------------------------------
__global__ void k_invert(float* __restrict__ G, float* __restrict__ Ginv) {
    __shared__ float pg[DIMD];
    __shared__ float pi[DIMD];
    int n = blockIdx.x;
    float* Gn = G    + (size_t)n * DIMD * DIMD;
    float* In = Ginv + (size_t)n * DIMD * DIMD;

    for (int idx = threadIdx.x; idx < DIMD * DIMD; idx += blockDim.x)
        In[idx] = ((idx >> 9) == (idx & 511)) ? 1.f : 0.f;
    __threadfence();
    __syncthreads();

    for (int p = 0; p < DIMD; ++p) {
        float piv = Gn[p * DIMD + p];
        float rcp = (fabsf(piv) > 1e-30f) ? 1.f / piv : 0.f;
        for (int c = threadIdx.x; c < DIMD; c += blockDim.x) {
            float gv = Gn[p * DIMD + c] * rcp;
            float iv = In[p * DIMD + c] * rcp;
            pg[c] = gv; pi[c] = iv;
            Gn[p * DIMD + c] = gv;
            In[p * DIMD + c] = iv;
        }
        __threadfence();
        __syncthreads();
#pragma unroll
        for (int rr = 0; rr < 2; ++rr) {
            int r = threadIdx.x + rr * 256;
            if (r == p) continue;
            float f = Gn[r * DIMD + p];
            if (f != 0.f) {
                for (int c = 0; c < DIMD; ++c) {
                    Gn[r * DIMD + c] -= f * pg[c];
                    In[r * DIMD + c] -= f * pi[c];
                }
            }
        }
        __threadfence();
        __syncthreads();
    }
}

// ---------------------------------------------------------------------------
// Kernel 7: trans[n] = STT[n] @ Ginv[n]  (512x512x512 per n) via WMMA -> d_out
// ---------------------------------------------------------------------------
__global__ void k_final(const float* __restrict__ STT,
                        const float* __restrict__ Ginv,
                        float* __restrict__ out) {
    int n = blockIdx.x >> 7, tg = blockIdx.x & 127;
    const int lane = threadIdx.x & 31;
    const int wave = threadIdx.x >> 5;
    const int t = tg * 8 + wave;
    const int d0 = (t >> 5) << 4, e0 = (t & 31) << 4;
    const int m = lane & 15;
    const int kkb = (lane >> 4) << 1;
    size_t base = (size_t)n * DIMD * DIMD;

    v8f acc = {};
    for (int c0 = 0; c0 < DIMD; c0 += 4) {
        v2f a = *(const v2f*)&STT[base + (d0 + m) * DIMD + c0 + kkb];
        v2f b;
        b.x = Ginv[base + (c0 + kkb) * DIMD + e0 + m];
        b.y = Ginv[base + (c0 + kkb + 1) * DIMD + e0 + m];
        acc = wmma4(a, b, acc);
    }
#pragma unroll
    for (int rr = 0; rr < 8; ++rr) {
        int row = d0 + rr + ((lane >> 4) << 3);
        out[base + row * DIMD + e0 + m] = acc[rr];
    }
}

// ---------------------------------------------------------------------------
extern "C" void kernel_launch(void* const* d_in, const int* in_sizes, int n_in,
                              void* d_out, int out_size, void* d_ws, size_t ws_size,
                              hipStream_t stream) {
    (void)in_sizes; (void)n_in; (void)out_size; (void)ws_size;
    const float* sfeat = (const float*)d_in[0];
    const float* tfeat = (const float*)d_in[1];
    // d_in[2], d_in[3] (labels) are unused by the reference computation.

    float* ws   = (float*)d_ws;
    float* cs   = ws;                          // [8][4][64][512]  (ct follows)
    float* STT  = ws + 2097152;                // [8][512][512]
    float* G    = ws + 4194304;                // [8][512][512]
    float* Ginv = ws + 6291456;                // [8][512][512]
    int*   assign = (int*)(ws + 8388608);      // 64 * 1024
    int*   y      = (int*)(ws + 8454144);      // 8 * 4 * 4 * 64
    float* out = (float*)d_out;                // [8][512][512]

    k_init_centers<<<64, 256, 0, stream>>>(sfeat, tfeat, cs);
    size_t upd_lds = (size_t)(KC * DIMD + KC) * sizeof(float); // ~128 KB
    for (int it = 0; it < KM_ITERS; ++it) {
        k_assign<<<64, 256, 0, stream>>>(sfeat, tfeat, cs, assign);
        k_update<<<64, 256, upd_lds, stream>>>(sfeat, tfeat, assign, cs);
    }
    k_crossassign<<<128, 128, 0, stream>>>(cs, cs + 1048576, y);
    k_gram<<<1024, 256, 0, stream>>>(cs, cs + 1048576, y, STT, G);
    k_invert<<<8, 256, 0, stream>>>(G, Ginv);
    k_final<<<1024, 256, 0, stream>>>(STT, Ginv, out);
}